// FeatureExtractor_59072980189354
// MI455X (gfx1250) — compile-verified
//
#include <hip/hip_runtime.h>
#include <stdint.h>

#define INPUT_SIZE 16384
#define NUM_CONV   128
#define KMAX       11
#define BATCH      256
#define THREADS    256
#define WAVES      (THREADS / 32)
#define CONV_SPLIT 8                              // blocks per batch row
#define CONV_PER_BLOCK (NUM_CONV / CONV_SPLIT)    // 16
#define CONV_PER_WAVE  (CONV_PER_BLOCK / WAVES)   // 2
#define NC_MAX     10                             // max K-chunks (Ktot<=40) for WMMA path

typedef __attribute__((ext_vector_type(4))) unsigned int v4u;
typedef __attribute__((ext_vector_type(8))) int          v8i;
typedef __attribute__((ext_vector_type(4))) int          v4i;
typedef __attribute__((ext_vector_type(2))) float        v2f;
typedef __attribute__((ext_vector_type(8))) float        v8f;

// ---------------- VALU path (any dilation) ----------------
template <int K>
__device__ __forceinline__ float taps_unchecked(const float* __restrict__ sm,
                                                const float* __restrict__ w,
                                                float acc, int base, int d) {
#pragma unroll
    for (int j = 0; j < K; ++j)
        acc = fmaf(w[j], sm[base + j * d], acc);
    return acc;
}

template <int K>
__device__ __forceinline__ float taps_checked(const float* __restrict__ sm,
                                              const float* __restrict__ w,
                                              float acc, int base, int d) {
#pragma unroll
    for (int j = 0; j < K; ++j) {
        const int idx = base + j * d;
        if ((unsigned)idx < (unsigned)INPUT_SIZE)
            acc = fmaf(w[j], sm[idx], acc);
    }
    return acc;
}

template <int K>
__device__ __forceinline__ void conv_one(const float* __restrict__ sm,
                                         const float* __restrict__ w,
                                         float bi, int d, int p, int L, int lane,
                                         float& mxOut, int& cntOut) {
    const int tLo = p;
    const int tHi = (INPUT_SIZE - 1) - (K - 1) * d + p;

    float mx = -__builtin_inff();
    int   cnt = 0;

    int t = lane;
    for (; t + 32 < L; t += 64) {                 // 2 outputs in flight per lane
        const int t1 = t + 32;
        float a0 = bi, a1 = bi;
        if (t >= tLo && t1 <= tHi) {
            a0 = taps_unchecked<K>(sm, w, a0, t  - p, d);
            a1 = taps_unchecked<K>(sm, w, a1, t1 - p, d);
        } else {
            a0 = taps_checked<K>(sm, w, a0, t  - p, d);
            a1 = taps_checked<K>(sm, w, a1, t1 - p, d);
        }
        mx = fmaxf(mx, fmaxf(a0, a1));
        cnt += (a0 >= 0.0f) ? 1 : 0;
        cnt += (a1 >= 0.0f) ? 1 : 0;
    }
    for (; t < L; t += 32) {
        float a = bi;
        if (t >= tLo && t <= tHi) a = taps_unchecked<K>(sm, w, a, t - p, d);
        else                      a = taps_checked<K>(sm, w, a, t - p, d);
        mx = fmaxf(mx, a);
        cnt += (a >= 0.0f) ? 1 : 0;
    }
    mxOut = mx;
    cntOut = cnt;
}

// ---------------- WMMA Toeplitz path (small dilation) ----------------
// D[m,n] = y[t0+m+16n] = sum_k A[m,k]*B[k,n],  NC = ceil((16+(k-1)d)/4) chunks.
// A[m,k] = w[(k-m)/d] iff (k-m) is a non-negative multiple of d, else 0.
// B[k,n] = x[t0-p+k+16n] with zero for out-of-range (conv zero padding).
#if defined(__has_builtin) && __has_builtin(__builtin_amdgcn_wmma_f32_16x16x4_f32)
#define ROCKET_WMMA 1
template <int NC>
__device__ __forceinline__ void conv_wmma(const float* __restrict__ sm,
                                          const float* __restrict__ w,
                                          float bi, int d, int p, int L,
                                          int lane, float& mxOut, int& cntOut) {
    const int m  = lane & 15;     // A row / B col / D col index per lane
    const int kh = lane >> 4;     // K-half select (ISA A/B VGPR layout)

    // Build the Toeplitz A chunks once per conv (registers, static indexing).
    float a0[NC], a1[NC];
#pragma unroll
    for (int c = 0; c < NC; ++c) {
        float av0 = 0.0f, av1 = 0.0f;
        const int d0 = 4 * c + 2 * kh - m;        // kappa - m for r=0
#pragma unroll
        for (int j = 0; j < KMAX; ++j) {          // w[] zero-padded beyond k
            av0 = (d0     == j * d) ? w[j] : av0;
            av1 = (d0 + 1 == j * d) ? w[j] : av1;
        }
        a0[c] = av0;
        a1[c] = av1;
    }

    const int offl = 2 * kh + 16 * m;             // per-lane B gather offset
    const float NEG_INF = -__builtin_inff();
    float mx = NEG_INF;
    int   cnt = 0;

    for (int t0 = 0; t0 < L; t0 += 256) {         // 256 outputs per 16x16 tile
        const int base0 = t0 - p + offl;

        // Phase 1: batched B gathers (one dscnt wait instead of one per chunk)
        float b0[NC], b1[NC];
#pragma unroll
        for (int c = 0; c < NC; ++c) {
            const int i0 = base0 + 4 * c;
            const int i1 = i0 + 1;
            const bool ok0 = (unsigned)i0 < (unsigned)INPUT_SIZE;
            const bool ok1 = (unsigned)i1 < (unsigned)INPUT_SIZE;
            float v0 = sm[ok0 ? i0 : 0];
            float v1 = sm[ok1 ? i1 : 0];
            b0[c] = ok0 ? v0 : 0.0f;
            b1[c] = ok1 ? v1 : 0.0f;
        }

        // Phase 2: back-to-back WMMA accumulation chain
        v8f acc = {};
#pragma unroll
        for (int c = 0; c < NC; ++c) {
            v2f av = { a0[c], a1[c] };
            v2f bv = { b0[c], b1[c] };
            acc = __builtin_amdgcn_wmma_f32_16x16x4_f32(
                false, av, false, bv, (short)0, acc, false, false);
        }

        // Phase 3: branchless epilogue. Lane holds D[M=r+8*kh, N=m]:
        // t = t0 + M + 16*N; mask tail with selects, not exec branches.
#pragma unroll
        for (int r = 0; r < 8; ++r) {
            const int  t     = t0 + r + 8 * kh + 16 * m;
            const bool valid = (t < L);
            const float v    = acc[r] + bi;
            mx = fmaxf(mx, valid ? v : NEG_INF);
            cnt += (valid && (v >= 0.0f)) ? 1 : 0;
        }
    }
    mxOut = mx;
    cntOut = cnt;
}
#else
#define ROCKET_WMMA 0
#endif

__global__ __launch_bounds__(THREADS) void rocket_feat_kernel(
    const float* __restrict__ x, const float* __restrict__ W,
    const float* __restrict__ bias, const int* __restrict__ ks,
    const int* __restrict__ dil, const int* __restrict__ pad,
    float* __restrict__ out)
{
    extern __shared__ float sm[];                 // 16384 f32 = 64 KB: one batch row
    const int b   = blockIdx.x;
    const int tid = threadIdx.x;
    const float* row = x + (size_t)b * INPUT_SIZE;

    // ---- Kick off the row DMA into LDS via the Tensor Data Mover ----
#if defined(__has_builtin) && __has_builtin(__builtin_amdgcn_tensor_load_to_lds)
#define ROCKET_TDM 1
    if (tid == 0) {   // single wave issues the DMA; TDM ignores EXEC
        const uint64_t ga = (uint64_t)(uintptr_t)row;
        const uint32_t la = (uint32_t)(uintptr_t)(&sm[0]);
        v4u g0 = { 1u,                                      // count=1
                   la,                                      // lds_addr
                   (unsigned)(ga & 0xffffffffu),            // global_addr lo
                   (unsigned)((ga >> 32) & 0x1ffffffu) | (2u << 30) }; // hi|type=2
        v8i g1 = { (int)0x00020000,   // wg_mask=0, data_size=2 (4B)
                   (int)0x40000000,   // tensor_dim0[15:0]=0x4000 @ bits63:48
                   (int)0x00010000,   // tensor_dim1=1            @ bits95:80
                   (int)0x40000000,   // tile_dim0=0x4000         @ bits127:112
                   1,                 // tile_dim1=1
                   INPUT_SIZE,        // tensor_dim0_stride lo
                   0, 0 };
        v4i gz = { 0, 0, 0, 0 };
#if __clang_major__ >= 23
        v8i gz8 = { 0, 0, 0, 0, 0, 0, 0, 0 };
        __builtin_amdgcn_tensor_load_to_lds(g0, g1, gz, gz, gz8, 0);
#else
        __builtin_amdgcn_tensor_load_to_lds(g0, g1, gz, gz, 0);
#endif
    }
#else
#define ROCKET_TDM 0
#endif

    // ---- Overlap the DMA with wave-uniform parameter/weight preloads ----
    const int wave  = tid >> 5;
    const int lane  = tid & 31;
    const int conv0 = blockIdx.y * CONV_PER_BLOCK + wave * CONV_PER_WAVE;

    int   kk[CONV_PER_WAVE], dd[CONV_PER_WAVE], pp[CONV_PER_WAVE], LL[CONV_PER_WAVE];
    float bb[CONV_PER_WAVE];
    float ww[CONV_PER_WAVE][KMAX];
#pragma unroll
    for (int ci = 0; ci < CONV_PER_WAVE; ++ci) {
        const int i = conv0 + ci;
        kk[ci] = ks[i];
        dd[ci] = dil[i];
        pp[ci] = pad[i];
        LL[ci] = INPUT_SIZE + 2 * pp[ci] - dd[ci] * (kk[ci] - 1);
        bb[ci] = bias[i];
#pragma unroll
        for (int j = 0; j < KMAX; ++j) ww[ci][j] = W[i * KMAX + j]; // zero-padded
    }

#if ROCKET_TDM
    if (tid == 0) {
#if __has_builtin(__builtin_amdgcn_s_wait_tensorcnt)
        __builtin_amdgcn_s_wait_tensorcnt(0);
#else
        asm volatile("s_wait_tensorcnt 0x0" ::: "memory");
#endif
    }
    __syncthreads();
#else
    for (int v = tid; v < INPUT_SIZE / 4; v += THREADS)
        ((float4*)sm)[v] = ((const float4*)row)[v];
    __syncthreads();
#endif

    // ---- Each wave privately sweeps its convs (no further barriers) ----
    for (int ci = 0; ci < CONV_PER_WAVE; ++ci) {
        const int i = conv0 + ci;
        const int k = kk[ci], d = dd[ci], p = pp[ci], L = LL[ci];
        const float bi = bb[ci];
        const float* w = ww[ci];

        float mx;
        int cnt;
#if ROCKET_WMMA
        const int Ktot = 16 + (k - 1) * d;        // Toeplitz inner length
        if (Ktot <= 4 * NC_MAX) {
            // dense-enough Toeplitz: matrix-core path, NC fixed at compile time
            switch ((Ktot + 3) >> 2) {
                case 6:  conv_wmma<6 >(sm, w, bi, d, p, L, lane, mx, cnt); break;
                case 7:  conv_wmma<7 >(sm, w, bi, d, p, L, lane, mx, cnt); break;
                case 8:  conv_wmma<8 >(sm, w, bi, d, p, L, lane, mx, cnt); break;
                case 9:  conv_wmma<9 >(sm, w, bi, d, p, L, lane, mx, cnt); break;
                default: conv_wmma<10>(sm, w, bi, d, p, L, lane, mx, cnt); break;
            }
        } else
#endif
        {
            if (k == 7)      conv_one<7 >(sm, w, bi, d, p, L, lane, mx, cnt);
            else if (k == 9) conv_one<9 >(sm, w, bi, d, p, L, lane, mx, cnt);
            else             conv_one<11>(sm, w, bi, d, p, L, lane, mx, cnt);
        }

        // wave32 reduction: max + count
#pragma unroll
        for (int off = 16; off; off >>= 1) {
            mx = fmaxf(mx, __shfl_xor(mx, off, 32));
            cnt += __shfl_xor(cnt, off, 32);
        }
        if (lane == 0) {
            const size_t o = (size_t)b * (2 * NUM_CONV) + 2 * (size_t)i;
            out[o]     = mx;
            out[o + 1] = (float)cnt / (float)L;
        }
    }
}

extern "C" void kernel_launch(void* const* d_in, const int* in_sizes, int n_in,
                              void* d_out, int out_size, void* d_ws, size_t ws_size,
                              hipStream_t stream) {
    (void)in_sizes; (void)n_in; (void)out_size; (void)d_ws; (void)ws_size;
    const float* x    = (const float*)d_in[0];
    const float* W    = (const float*)d_in[1];
    const float* bias = (const float*)d_in[2];
    const int*   ks   = (const int*)d_in[3];
    const int*   dil  = (const int*)d_in[4];
    const int*   pad  = (const int*)d_in[5];
    float* out = (float*)d_out;

    dim3 grid(BATCH, CONV_SPLIT);
    rocket_feat_kernel<<<grid, THREADS, INPUT_SIZE * sizeof(float), stream>>>(
        x, W, bias, ks, dil, pad, out);
}